// Renderer_56401510531596
// MI455X (gfx1250) — compile-verified
//
#include <hip/hip_runtime.h>
#include <hip/hip_bf16.h>

typedef __attribute__((ext_vector_type(2))) float v2f;
typedef __attribute__((ext_vector_type(8))) float v8f;

#define HH 256
#define WW 256
#define NV 8000
#define NT 16000
#define TCHUNK 128
#define EPSF 1e-9f

// SH constants (double-precision products, cast to f32, matching the reference's
// trace-time constant folding)
#define KC_Y0 0.8862269254527580137f   /* A0*C0 = pi/sqrt(4pi)          */
#define K1_C  1.7724538509055160273f   /* A1*C1 = sqrt(pi)              */
#define K2_C  2.4270323905f            /* A2*C2                          */
#define K6_C  0.7006240352f            /* A2*C2*0.5/sqrt(3)              */
#define K8_C  1.2135161953f            /* A2*C2*0.5                      */

__device__ __forceinline__ float Gval(const float* __restrict__ gamma, int k, int n) {
    // g[k][n] for the padded 12x16 B matrix: zero outside the real 9x3 block.
    if (n >= 3 || k >= 9) return 0.0f;
    float v = gamma[k * 3 + n];
    if (k == 0) v += 0.7f;   // init_light adds 0.7 to row 0, all channels
    return v;
}

// ---------------------------------------------------------------------------
// Kernel 1: SH shading via V_WMMA_F32_16X16X4_F32.
// One wave (32 lanes) per 16-vertex tile. lighting = Y(16x9) @ g(9x3),
// K padded to 12 -> three chained 16x16x4 WMMAs. colors = diffuse * lighting.
// ---------------------------------------------------------------------------
__global__ __launch_bounds__(32)
void sh_shade_wmma(const float* __restrict__ normals,
                   const float* __restrict__ diffuse,
                   const float* __restrict__ gamma,
                   float* __restrict__ colors)
{
    const int tile = blockIdx.x;        // 0..499
    const int lane = threadIdx.x;       // 0..31
    const int m    = lane & 15;
    const int vtx  = tile * 16 + m;     // NV == 500*16 exactly

    const float nx = normals[vtx * 3 + 0];
    const float ny = normals[vtx * 3 + 1];
    const float nz = normals[vtx * 3 + 2];

    const float Y0v = KC_Y0;
    const float Y1v = -K1_C * ny;
    const float Y2v =  K1_C * nz;
    const float Y3v = -K1_C * nx;
    const float Y4v =  K2_C * nx * ny;
    const float Y5v = -K2_C * ny * nz;
    const float Y6v =  K6_C * (3.0f * nz * nz - 1.0f);
    const float Y7v = -K2_C * nx * nz;
    const float Y8v =  K8_C * (nx * nx - ny * ny);

    const bool lo = (lane < 16);        // lane half selects K pair per ISA layout
    const int  n  = lane & 15;          // B/N column (output channel)
    const int  kb = lo ? 0 : 2;

    // A fragments (16x4 f32): VGPR0 = K=kb, VGPR1 = K=kb+1, M = lane%16
    v2f a0 = { lo ? Y0v : Y2v, lo ? Y1v : Y3v };   // K 0..3
    v2f a1 = { lo ? Y4v : Y6v, lo ? Y5v : Y7v };   // K 4..7
    v2f a2 = { lo ? Y8v : 0.0f, 0.0f };            // K 8..11 (9..11 zero pad)

    // B fragments (4x16 f32): VGPR0 = K=kb, VGPR1 = K=kb+1, N = lane%16
    v2f b0 = { Gval(gamma, kb + 0, n), Gval(gamma, kb + 1, n) };
    v2f b1 = { Gval(gamma, kb + 4, n), Gval(gamma, kb + 5, n) };
    v2f b2 = { Gval(gamma, kb + 8, n), Gval(gamma, kb + 9, n) };

    v8f acc = {};
    acc = __builtin_amdgcn_wmma_f32_16x16x4_f32(false, a0, false, b0, (short)0, acc, false, false);
    acc = __builtin_amdgcn_wmma_f32_16x16x4_f32(false, a1, false, b1, (short)0, acc, false, false);
    acc = __builtin_amdgcn_wmma_f32_16x16x4_f32(false, a2, false, b2, (short)0, acc, false, false);

    // D layout: VGPR r -> M = r (lanes 0-15) or r+8 (lanes 16-31), N = lane%16
    if (n < 3) {
        const int mbase = lo ? 0 : 8;
#pragma unroll
        for (int r = 0; r < 8; ++r) {
            const int vm = tile * 16 + mbase + r;
            colors[vm * 3 + n] = diffuse[vm * 3 + n] * acc[r];
        }
    }
}

// ---------------------------------------------------------------------------
// Kernel 2: per-triangle setup, SoA (12 fields x NT).
// Degenerate triangles get NaN-poisoned edge terms so every barycentric
// comparison fails (matches the reference's `good` mask semantics).
// ---------------------------------------------------------------------------
__global__ __launch_bounds__(256)
void tri_setup(const float* __restrict__ verts,
               const int* __restrict__ tris,
               float* __restrict__ S)
{
    const int t = blockIdx.x * blockDim.x + threadIdx.x;
    if (t >= NT) return;

    float xs[3], ys[3], zs[3], iws[3];
#pragma unroll
    for (int j = 0; j < 3; ++j) {
        const int i  = tris[t * 3 + j];
        const float vx = verts[i * 4 + 0];
        const float vy = verts[i * 4 + 1];
        const float vz = verts[i * 4 + 2];
        const float vw = verts[i * 4 + 3];
        const float iw = 1.0f / vw;
        xs[j]  = (vx * iw * 0.5f + 0.5f) * (float)(WW - 1);
        ys[j]  = (0.5f - 0.5f * vy * iw) * (float)(HH - 1);
        zs[j]  = vz * iw;
        iws[j] = iw;
    }

    const float d    = (ys[1] - ys[2]) * (xs[0] - xs[2]) + (xs[2] - xs[1]) * (ys[0] - ys[2]);
    const bool  good = fabsf(d) > EPSF;
    const float dinv = good ? (1.0f / d) : __builtin_nanf("");

    S[ 0 * NT + t] = xs[2];
    S[ 1 * NT + t] = ys[2];
    S[ 2 * NT + t] = (ys[1] - ys[2]) * dinv;
    S[ 3 * NT + t] = (xs[2] - xs[1]) * dinv;
    S[ 4 * NT + t] = (ys[2] - ys[0]) * dinv;
    S[ 5 * NT + t] = (xs[0] - xs[2]) * dinv;
    S[ 6 * NT + t] = zs[0];
    S[ 7 * NT + t] = zs[1];
    S[ 8 * NT + t] = zs[2];
    S[ 9 * NT + t] = iws[0];
    S[10 * NT + t] = iws[1];
    S[11 * NT + t] = iws[2];
}

// ---------------------------------------------------------------------------
// Kernel 3: z-buffer rasterizer. 256 pixels/block; 128-triangle chunks staged
// in LDS (6 KB), broadcast-read by all lanes. Strict `<` + in-order iteration
// reproduces the reference's tie-breaking exactly. Color fetch deferred.
// ---------------------------------------------------------------------------
__global__ __launch_bounds__(256)
void raster(const float* __restrict__ S,
            const int* __restrict__ tris,
            const float* __restrict__ colors,
            float* __restrict__ out)
{
    __shared__ float lds[12 * TCHUNK];

    const int   p  = blockIdx.x * 256 + threadIdx.x;   // pixel = y*W + x
    const float px = (float)(p % WW);
    const float py = (float)(p / WW);

    float bestD = __builtin_inff();
    int   bestT = -1;
    float bp0 = 0.0f, bp1 = 0.0f, bp2 = 0.0f;

    for (int c = 0; c < NT / TCHUNK; ++c) {
        __syncthreads();
        for (int i = threadIdx.x; i < 12 * TCHUNK; i += 256) {
            const int f  = i / TCHUNK;
            const int tt = i - f * TCHUNK;
            lds[i] = S[f * NT + c * TCHUNK + tt];
        }
        __syncthreads();

        for (int j = 0; j < TCHUNK; ++j) {
            const float x2  = lds[0 * TCHUNK + j];
            const float y2  = lds[1 * TCHUNK + j];
            const float e0x = lds[2 * TCHUNK + j];
            const float e0y = lds[3 * TCHUNK + j];
            const float e1x = lds[4 * TCHUNK + j];
            const float e1y = lds[5 * TCHUNK + j];
            const float dx = px - x2;
            const float dy = py - y2;
            const float b0 = e0x * dx + e0y * dy;   // NaN if degenerate -> excluded
            const float b1 = e1x * dx + e1y * dy;
            const float b2 = 1.0f - b0 - b1;
            if (b0 >= 0.0f && b1 >= 0.0f && b2 >= 0.0f) {
                const float bw0 = b0 * lds[ 9 * TCHUNK + j];
                const float bw1 = b1 * lds[10 * TCHUNK + j];
                const float bw2 = b2 * lds[11 * TCHUNK + j];
                float denom = bw0 + bw1 + bw2;
                denom = (fabsf(denom) > EPSF) ? denom : 1.0f;
                const float rd = 1.0f / denom;
                const float p0 = bw0 * rd;
                const float p1 = bw1 * rd;
                const float p2 = bw2 * rd;
                const float depth = p0 * lds[6 * TCHUNK + j]
                                  + p1 * lds[7 * TCHUNK + j]
                                  + p2 * lds[8 * TCHUNK + j];
                if (depth < bestD) {
                    bestD = depth;
                    bestT = c * TCHUNK + j;
                    bp0 = p0; bp1 = p1; bp2 = p2;
                }
            }
        }
    }

    float r = 0.0f, g = 0.0f, b = 0.0f, alpha = 0.0f;
    if (bestT >= 0) {
        const int i0 = tris[bestT * 3 + 0];
        const int i1 = tris[bestT * 3 + 1];
        const int i2 = tris[bestT * 3 + 2];
        r = bp0 * colors[i0 * 3 + 0] + bp1 * colors[i1 * 3 + 0] + bp2 * colors[i2 * 3 + 0];
        g = bp0 * colors[i0 * 3 + 1] + bp1 * colors[i1 * 3 + 1] + bp2 * colors[i2 * 3 + 1];
        b = bp0 * colors[i0 * 3 + 2] + bp1 * colors[i1 * 3 + 2] + bp2 * colors[i2 * 3 + 2];
        alpha = 1.0f;
    }
    out[p * 3 + 0] = r;
    out[p * 3 + 1] = g;
    out[p * 3 + 2] = b;
    out[HH * WW * 3 + p] = alpha;
}

// ---------------------------------------------------------------------------
// Launch wrapper. Inputs (setup_inputs order):
//   0 clip_vertices f32[1,8000,4]   1 triangles i32[16000,3]
//   2 normals f32[1,8000,3]         3 diffuse_colors f32[1,8000,3]
//   4 gamma f32[1,27]
// Output: image f32[1,256,256,3] ++ alpha f32[1,256,256,1]
// Workspace: colors (24000 f32) | tri setup SoA (12*16000 f32)
// ---------------------------------------------------------------------------
extern "C" void kernel_launch(void* const* d_in, const int* in_sizes, int n_in,
                              void* d_out, int out_size, void* d_ws, size_t ws_size,
                              hipStream_t stream) {
    const float* clipv   = (const float*)d_in[0];
    const int*   tris    = (const int*)d_in[1];
    const float* normals = (const float*)d_in[2];
    const float* diffuse = (const float*)d_in[3];
    const float* gamma   = (const float*)d_in[4];
    float* out = (float*)d_out;
    float* ws  = (float*)d_ws;

    float* colors = ws;             // NV*3 floats
    float* S      = ws + NV * 3;    // 12*NT floats

    sh_shade_wmma<<<NV / 16, 32, 0, stream>>>(normals, diffuse, gamma, colors);
    tri_setup<<<(NT + 255) / 256, 256, 0, stream>>>(clipv, tris, S);
    raster<<<(HH * WW) / 256, 256, 0, stream>>>(S, tris, colors, out);
}